// SimpleGATLayer_2173253452571
// MI455X (gfx1250) — compile-verified
//
#include <hip/hip_runtime.h>
#include <hip/hip_bf16.h>

#define IN_DIM  128        // also HEADS*OUT_DIM
#define HEADS   2
#define OUT_DIM 64

typedef __attribute__((ext_vector_type(2))) float v2f;
typedef __attribute__((ext_vector_type(8))) float v8f;

// monotonic float <-> uint key for atomicMax on floats
__device__ __forceinline__ unsigned fkey(float f) {
    unsigned b = __float_as_uint(f);
    return (b & 0x80000000u) ? ~b : (b | 0x80000000u);
}
__device__ __forceinline__ float funkey(unsigned k) {
    unsigned b = (k & 0x80000000u) ? (k ^ 0x80000000u) : ~k;
    return __uint_as_float(b);
}

// ---------------- 1. init: zero output, reset reduction scalars ----------------
__global__ void gat_init(float4* __restrict__ out4, long long n4,
                         unsigned* __restrict__ maxkey, float* __restrict__ sums) {
    long long i = (long long)blockIdx.x * blockDim.x + threadIdx.x;
    if (i < n4) out4[i] = make_float4(0.f, 0.f, 0.f, 0.f);
    if (i == 0) {
        maxkey[0] = 0u; maxkey[1] = 0u;   // encodes -inf
        sums[0] = 0.f;  sums[1] = 0.f;
    }
}

// ---------------- 2. h = x @ W via V_WMMA_F32_16X16X4_F32 ----------------
// one wave computes one 16x16 tile of h; 8 waves per block; K=128 -> 32 WMMAs
__global__ void __launch_bounds__(256) gat_gemm_wmma(
        const float* __restrict__ x, const float* __restrict__ W,
        float* __restrict__ h, int N) {
    const int lane  = threadIdx.x & 31;
    const int wv    = threadIdx.x >> 5;
    const int tile  = blockIdx.x * 8 + wv;
    const int colT  = IN_DIM / 16;                  // 8 column tiles
    const int rt    = tile / colT;
    const int col0  = (tile % colT) * 16;
    if (rt * 16 >= N) return;                       // wave-uniform exit

    const int m    = lane & 15;                     // row (A) / col (B) within tile
    const int ksel = lane >> 4;                     // lanes 16-31 hold K+2..K+3

    int row  = rt * 16 + m;
    int rowc = row < N ? row : N - 1;               // clamp tail-tile loads
    const float* xrow = x + (size_t)rowc * IN_DIM;

    v8f acc = {};
#pragma unroll
    for (int kk = 0; kk < IN_DIM / 4; ++kk) {
        const int kb = kk * 4 + ksel * 2;
        // A: 16x4 fp32 fragment — lane holds A[m][kb], A[m][kb+1]
        v2f a = *(const v2f*)(xrow + kb);
        // B: 4x16 fp32 fragment — lane holds W[kb][col0+m], W[kb+1][col0+m]
        v2f b;
        b.x = W[(size_t)kb       * IN_DIM + col0 + m];
        b.y = W[(size_t)(kb + 1) * IN_DIM + col0 + m];
        acc = __builtin_amdgcn_wmma_f32_16x16x4_f32(
                /*neg_a=*/false, a, /*neg_b=*/false, b,
                /*c_mod=*/(short)0, acc, /*reuse_a=*/false, /*reuse_b=*/false);
    }
    // C/D layout: VGPR v -> (M = v + ksel*8, N = m)
#pragma unroll
    for (int v = 0; v < 8; ++v) {
        int r = rt * 16 + v + ksel * 8;
        if (r < N) h[(size_t)r * IN_DIM + col0 + m] = acc[v];
    }
}

// ---------------- 3. per-node attention scalars al / ar ----------------
// one wave per node; lane covers 4 channels; half-wave = one head
__global__ void __launch_bounds__(256) gat_alar(
        const float* __restrict__ h, const float* __restrict__ attn_l,
        const float* __restrict__ attn_r, float* __restrict__ al,
        float* __restrict__ ar, int N) {
    const int lane = threadIdx.x & 31;
    const int node = blockIdx.x * 8 + (threadIdx.x >> 5);
    if (node >= N) return;
    const int head = lane >> 4;            // lanes 0-15: head0, 16-31: head1
    const int d4   = (lane & 15) * 4;

    float4 hv = *(const float4*)(h + (size_t)node * IN_DIM + head * OUT_DIM + d4);
    float4 lv = *(const float4*)(attn_l + head * OUT_DIM + d4);
    float4 rv = *(const float4*)(attn_r + head * OUT_DIM + d4);
    float pl = hv.x * lv.x + hv.y * lv.y + hv.z * lv.z + hv.w * lv.w;
    float pr = hv.x * rv.x + hv.y * rv.y + hv.z * rv.z + hv.w * rv.w;
#pragma unroll
    for (int msk = 1; msk < 16; msk <<= 1) {
        pl += __shfl_xor(pl, msk, 32);
        pr += __shfl_xor(pr, msk, 32);
    }
    if ((lane & 15) == 0) {
        al[node * HEADS + head] = pl;
        ar[node * HEADS + head] = pr;
    }
}

// ---------------- 4. alpha = leaky_relu(al[src]+ar[dst]); global max/head ----------------
__global__ void __launch_bounds__(256) gat_alpha_max(
        const int* __restrict__ src, const int* __restrict__ dst,
        const float* __restrict__ al, const float* __restrict__ ar,
        float* __restrict__ av, unsigned* __restrict__ maxkey, int E) {
    __shared__ float smax[8][2];
    long long i = (long long)blockIdx.x * 256 + threadIdx.x;
    const int lane = threadIdx.x & 31, wv = threadIdx.x >> 5;
    float a = -3.402823466e38f;
    if (i < (long long)E * HEADS) {
        int e = (int)(i >> 1), hd = (int)(i & 1);
        float v = al[(size_t)src[e] * HEADS + hd] + ar[(size_t)dst[e] * HEADS + hd];
        a = v > 0.f ? v : 0.2f * v;               // leaky_relu(0.2)
        av[i] = a;
    }
#pragma unroll
    for (int msk = 2; msk < 32; msk <<= 1) a = fmaxf(a, __shfl_xor(a, msk, 32));
    if (lane < 2) smax[wv][lane] = a;             // lane parity == head
    __syncthreads();
    if (threadIdx.x < 16) {
        float v = smax[threadIdx.x >> 1][threadIdx.x & 1];
#pragma unroll
        for (int msk = 2; msk < 16; msk <<= 1) v = fmaxf(v, __shfl_xor(v, msk, 32));
        if (threadIdx.x < 2) atomicMax(&maxkey[threadIdx.x], fkey(v));
    }
}

// ---------------- 5. expv = exp(alpha - max); global sum/head ----------------
__global__ void __launch_bounds__(256) gat_expsum(
        float* __restrict__ av, const unsigned* __restrict__ maxkey,
        float* __restrict__ sums, int E) {
    __shared__ float ssum[8][2];
    long long i = (long long)blockIdx.x * 256 + threadIdx.x;
    const int lane = threadIdx.x & 31, wv = threadIdx.x >> 5;
    float s = 0.f;
    if (i < (long long)E * HEADS) {
        int hd = (int)(i & 1);
        s = __expf(av[i] - funkey(maxkey[hd]));
        av[i] = s;                                 // overwrite alpha with exp value
    }
#pragma unroll
    for (int msk = 2; msk < 32; msk <<= 1) s += __shfl_xor(s, msk, 32);
    if (lane < 2) ssum[wv][lane] = s;
    __syncthreads();
    if (threadIdx.x < 16) {
        float v = ssum[threadIdx.x >> 1][threadIdx.x & 1];
#pragma unroll
        for (int msk = 2; msk < 16; msk <<= 1) v += __shfl_xor(v, msk, 32);
        if (threadIdx.x < 2) atomicAdd(&sums[threadIdx.x], v);
    }
}

// ---------------- 6. inv_sum ----------------
__global__ void gat_invsum(const float* __restrict__ sums, float* __restrict__ inv) {
    if (threadIdx.x < HEADS) inv[threadIdx.x] = 1.0f / sums[threadIdx.x];
}

// ---------------- 7. scatter: out[dst] += h[src] * w   (L2-resident atomics) ----------------
__global__ void __launch_bounds__(256) gat_scatter(
        const int* __restrict__ src, const int* __restrict__ dst,
        const float* __restrict__ h, const float* __restrict__ expv,
        const float* __restrict__ inv, float* __restrict__ out,
        long long total, int E) {
    long long i = (long long)blockIdx.x * 256 + threadIdx.x;
    if (i >= total) return;
    int e  = (int)(i >> 7);          // IN_DIM == 128 channels per edge
    int c  = (int)(i & 127);
    int hd = c >> 6;
    int s  = src[e], d = dst[e];
    float w   = expv[(size_t)e * HEADS + hd] * inv[hd];
    float val = h[(size_t)s * IN_DIM + c] * w;
    atomicAdd(&out[(size_t)d * IN_DIM + c], val);
}

extern "C" void kernel_launch(void* const* d_in, const int* in_sizes, int n_in,
                              void* d_out, int out_size, void* d_ws, size_t ws_size,
                              hipStream_t stream) {
    const float* x      = (const float*)d_in[0];
    const int*   eidx   = (const int*)  d_in[1];
    const float* W      = (const float*)d_in[2];
    const float* attn_l = (const float*)d_in[3];
    const float* attn_r = (const float*)d_in[4];
    float*       out    = (float*)d_out;

    const int N = in_sizes[0] / IN_DIM;
    const int E = in_sizes[1] / 2;
    const int* src = eidx;
    const int* dst = eidx + E;

    // workspace layout (256B-aligned slices)
    char* ws = (char*)d_ws;
    size_t off = 0;
    auto carve = [&](size_t bytes) -> void* {
        void* p = ws + off;
        off = (off + bytes + 255) & ~(size_t)255;
        return p;
    };
    float*    h      = (float*)   carve((size_t)N * IN_DIM * sizeof(float));   // 51.2 MB
    float*    al     = (float*)   carve((size_t)N * HEADS * sizeof(float));
    float*    ar     = (float*)   carve((size_t)N * HEADS * sizeof(float));
    float*    av     = (float*)   carve((size_t)E * HEADS * sizeof(float));    // alpha -> expv
    unsigned* maxkey = (unsigned*)carve(2 * sizeof(unsigned));
    float*    sums   = (float*)   carve(2 * sizeof(float));
    float*    inv    = (float*)   carve(2 * sizeof(float));
    (void)ws_size; (void)n_in; (void)out_size;

    // 1. zero output + reduction scalars
    long long n4 = (long long)N * IN_DIM / 4;
    gat_init<<<(unsigned)((n4 + 255) / 256), 256, 0, stream>>>((float4*)out, n4, maxkey, sums);

    // 2. WMMA GEMM
    int rowTiles = (N + 15) / 16;
    int tiles    = rowTiles * (IN_DIM / 16);
    gat_gemm_wmma<<<(unsigned)((tiles + 7) / 8), 256, 0, stream>>>(x, W, h, N);

    // 3. per-node attention scalars
    gat_alar<<<(unsigned)((N + 7) / 8), 256, 0, stream>>>(h, attn_l, attn_r, al, ar, N);

    // 4./5. softmax reductions over all edges per head
    long long eh = (long long)E * HEADS;
    gat_alpha_max<<<(unsigned)((eh + 255) / 256), 256, 0, stream>>>(src, dst, al, ar, av, maxkey, E);
    gat_expsum  <<<(unsigned)((eh + 255) / 256), 256, 0, stream>>>(av, maxkey, sums, E);
    gat_invsum  <<<1, 32, 0, stream>>>(sums, inv);

    // 7. weighted scatter-add
    long long total = (long long)E * IN_DIM;
    gat_scatter<<<(unsigned)((total + 255) / 256), 256, 0, stream>>>(src, dst, h, av, inv, out, total, E);
}